// MultiHeadAttention_76338748719525
// MI455X (gfx1250) — compile-verified
//
#include <hip/hip_runtime.h>
#include <hip/hip_bf16.h>

typedef __attribute__((ext_vector_type(2))) float v2f;
typedef __attribute__((ext_vector_type(8))) float v8f;

#define D_MODEL   1024
#define NUM_HEADS 16
#define HEAD_DIM  64
#define SEQ_T     1024
#define BATCH     2
#define M_ROWS    (BATCH * SEQ_T)   // 2048
#define SROW      1028              // padded LDS row stride (floats) for the score panel
#define VROW      68                // padded LDS row stride (floats) for the V chunk
#define VCHUNK    128               // V rows staged per iteration (VCHUNK*VROW*4 = 34,816 B)

__device__ __forceinline__ v8f wmma_f32x4(v2f a, v2f b, v8f c) {
  // D = A(16x4,f32) * B(4x16,f32) + C(16x16,f32)
  return __builtin_amdgcn_wmma_f32_16x16x4_f32(false, a, false, b, (short)0, c, false, false);
}

// Y = X @ W^T + bias.  X:[M_ROWS, D_MODEL], W:[D_MODEL, D_MODEL] (row n = weights of out-col n)
// SPLIT_HEADS: store Y as [B, H, T, hd]; else plain [M, N].
template <bool SPLIT_HEADS>
__global__ void gemm_bias_kernel(const float* __restrict__ X,
                                 const float* __restrict__ W,
                                 const float* __restrict__ bias,
                                 float* __restrict__ Y) {
  const int K = D_MODEL;
  int wave = threadIdx.x >> 5;
  int lane = threadIdx.x & 31;
  int half = lane >> 4;        // 0: K{0,1}/M{0..7}, 1: K{2,3}/M{8..15}
  int r    = lane & 15;

  int n0 = blockIdx.x * 64 + wave * 16;   // 4 waves cover 64 output cols
  int m0 = blockIdx.y * 16;

  const float* xrow = X + (size_t)(m0 + r) * K + 2 * half;
  const float* wrow = W + (size_t)(n0 + r) * K + 2 * half;

  // Two interleaved accumulator chains: loads of one chain overlap WMMA of the other.
  v8f acc0 = {}, acc1 = {};
  for (int k = 0; k < K; k += 8) {
    v2f a0 = *(const v2f*)(xrow + k);
    v2f b0 = *(const v2f*)(wrow + k);
    v2f a1 = *(const v2f*)(xrow + k + 4);
    v2f b1 = *(const v2f*)(wrow + k + 4);
    acc0 = wmma_f32x4(a0, b0, acc0);
    acc1 = wmma_f32x4(a1, b1, acc1);
  }
  v8f acc = acc0 + acc1;

  float bn = bias[n0 + r];
#pragma unroll
  for (int v = 0; v < 8; ++v) {
    int m = m0 + v + 8 * half;
    int n = n0 + r;
    float val = acc[v] + bn;
    if (SPLIT_HEADS) {
      int bb = m >> 10, t = m & 1023;
      int h = n >> 6, d = n & 63;
      Y[(((size_t)(bb * NUM_HEADS + h) * SEQ_T + t) << 6) + d] = val;
    } else {
      Y[(size_t)m * D_MODEL + n] = val;
    }
  }
}

// Fused: S = QK^T/8 -> 6-tick ODE -> +S_init -> softmax -> probs out + P@V
// One workgroup (256 thr = 8 waves) owns a 16-row score panel [16 x 1024] in LDS.
__global__ void attention_kernel(const float* __restrict__ Q,   // [B,H,T,hd]
                                 const float* __restrict__ Kt,  // [B,H,T,hd]
                                 const float* __restrict__ V,   // [B,H,T,hd]
                                 const float* __restrict__ tau_param,
                                 const float* __restrict__ w_memory,  // [H]
                                 float* __restrict__ probs,           // [B,H,T,T]
                                 float* __restrict__ attn_out) {      // [B,T,D]
  __shared__ float S[16 * SROW];            // 65,792 B
  __shared__ float Vbuf[VCHUNK * VROW];     // 34,816 B, async-staged V chunk
  __shared__ float red[256];

  int bh = blockIdx.y;
  int h  = bh & (NUM_HEADS - 1);
  int b  = bh >> 4;
  int i0 = blockIdx.x * 16;

  const float* Qp = Q + ((size_t)bh * SEQ_T + i0) * HEAD_DIM;
  const float* Kp = Kt + (size_t)bh * SEQ_T * HEAD_DIM;
  const float* Vp = V + (size_t)bh * SEQ_T * HEAD_DIM;

  int tid  = threadIdx.x;
  int wave = tid >> 5;
  int lane = tid & 31;
  int half = lane >> 4;
  int r    = lane & 15;

  // ---- Phase 1: S = Q K^T / 8 via WMMA; Q panel fragments held in registers ----
  v2f Af[16];
#pragma unroll
  for (int ks = 0; ks < 16; ++ks)
    Af[ks] = *(const v2f*)(Qp + r * HEAD_DIM + ks * 4 + 2 * half);

  for (int tj = 0; tj < 8; ++tj) {
    int j0 = (wave * 8 + tj) * 16;
    const float* kb = Kp + (size_t)(j0 + r) * HEAD_DIM + 2 * half;
    v8f acc = {};
#pragma unroll
    for (int ks = 0; ks < 16; ++ks) {
      v2f bfr = *(const v2f*)(kb + ks * 4);
      acc = wmma_f32x4(Af[ks], bfr, acc);
    }
#pragma unroll
    for (int v = 0; v < 8; ++v) {
      int m = v + 8 * half;
      S[m * SROW + j0 + r] = acc[v] * 0.125f;   // / sqrt(64)
    }
  }
  __syncthreads();

  // ---- Phase 2: tick dynamics + softmax (16 threads/row, 64 strided cols each) ----
  float tau     = log1pf(expf(tau_param[0]));   // softplus
  float inv_tau = 1.0f / tau;
  float wm      = w_memory[h];
  const float INV1 = 1.0f / (1.0f + 1e-6f);
  const float DT   = 1.0f / 6.0f;

  int row = tid >> 4;
  int c0  = tid & 15;
  float vals[64];
  float lmax = -3.4e38f;
  for (int c = 0; c < 64; ++c) {
    float s0 = S[row * SROW + c0 + 16 * c];
    float s  = s0;
#pragma unroll
    for (int t = 0; t < 6; ++t) {
      float ft = tanhf(s + s * INV1 * wm);      // tanh(s + sync*w_mem)
      float dh = ft - s * inv_tau;
      s = s + dh * DT;
      s = fminf(1.0f, fmaxf(-1.0f, s));
    }
    s += s0;                                    // + scores_init
    vals[c] = s;
    lmax = fmaxf(lmax, s);
  }
  red[tid] = lmax;
  __syncthreads();
  float rmax = red[row * 16];
#pragma unroll
  for (int k = 1; k < 16; ++k) rmax = fmaxf(rmax, red[row * 16 + k]);
  __syncthreads();

  float lsum = 0.0f;
  for (int c = 0; c < 64; ++c) {
    float e = expf(vals[c] - rmax);
    vals[c] = e;
    lsum += e;
  }
  red[tid] = lsum;
  __syncthreads();
  float rsum = 0.0f;
#pragma unroll
  for (int k = 0; k < 16; ++k) rsum += red[row * 16 + k];
  float inv = 1.0f / rsum;

  float* prow = probs + ((size_t)bh * SEQ_T + i0 + row) * SEQ_T;
  for (int c = 0; c < 64; ++c) {
    float p = vals[c] * inv;
    S[row * SROW + c0 + 16 * c] = p;   // keep P in LDS for P@V
    prow[c0 + 16 * c] = p;             // single HBM write of attn_probs
  }
  __syncthreads();

  // ---- Phase 3: O = P @ V. V staged in LDS by async copies; waves 0..3 own n-tiles ----
  int n0 = (wave & 3) * 16;
  v8f acc0 = {}, acc1 = {};

  for (int kc = 0; kc < SEQ_T; kc += VCHUNK) {
    // Cooperative async copy of V[kc .. kc+VCHUNK) x 64 (32 KB) into LDS with padded rows.
    // 2048 x 16B transfers, 8 per thread; tracked by ASYNCcnt.
    {
      const float* gsrc = Vp + (size_t)kc * HEAD_DIM;
#pragma unroll
      for (int it = 0; it < (VCHUNK * HEAD_DIM / 4) / 256; ++it) {
        int c16   = tid + it * 256;          // 16-byte chunk index
        int vrow  = c16 >> 4;                // V row within chunk (64 floats = 16 chunks)
        int vcol4 = c16 & 15;                // 4-float group within row
        unsigned ldsaddr = (unsigned)(size_t)&Vbuf[vrow * VROW + vcol4 * 4];
        const float* g = gsrc + vrow * HEAD_DIM + vcol4 * 4;
        asm volatile("global_load_async_to_lds_b128 %0, %1, off"
                     :: "v"(ldsaddr), "v"(g) : "memory");
      }
    }
    asm volatile("s_wait_asynccnt 0" ::: "memory");
    __syncthreads();

    if (wave < 4) {
      // Two interleaved WMMA chains over the staged chunk.
      for (int k = 0; k < VCHUNK; k += 8) {
        v2f a0 = *(const v2f*)(&S[r * SROW + kc + k + 2 * half]);
        v2f b0;
        b0.x = Vbuf[(k + 2 * half) * VROW + n0 + r];
        b0.y = Vbuf[(k + 2 * half + 1) * VROW + n0 + r];
        v2f a1 = *(const v2f*)(&S[r * SROW + kc + k + 4 + 2 * half]);
        v2f b1;
        b1.x = Vbuf[(k + 4 + 2 * half) * VROW + n0 + r];
        b1.y = Vbuf[(k + 4 + 2 * half + 1) * VROW + n0 + r];
        acc0 = wmma_f32x4(a0, b0, acc0);
        acc1 = wmma_f32x4(a1, b1, acc1);
      }
    }
    __syncthreads();   // protect Vbuf before next chunk's async copy
  }

  if (wave < 4) {
    v8f acc = acc0 + acc1;
#pragma unroll
    for (int v = 0; v < 8; ++v) {
      int m = v + 8 * half;
      attn_out[(size_t)(b * SEQ_T + i0 + m) * D_MODEL + h * HEAD_DIM + n0 + r] = acc[v];
    }
  }
}

extern "C" void kernel_launch(void* const* d_in, const int* in_sizes, int n_in,
                              void* d_out, int out_size, void* d_ws, size_t ws_size,
                              hipStream_t stream) {
  const float* query = (const float*)d_in[0];
  const float* key   = (const float*)d_in[1];
  const float* value = (const float*)d_in[2];
  const float* Wq    = (const float*)d_in[3];
  const float* bq    = (const float*)d_in[4];
  const float* Wk    = (const float*)d_in[5];
  const float* bk    = (const float*)d_in[6];
  const float* Wv    = (const float*)d_in[7];
  const float* bv    = (const float*)d_in[8];
  const float* Wo    = (const float*)d_in[9];
  const float* bo    = (const float*)d_in[10];
  const float* tau   = (const float*)d_in[11];
  /* d_in[12] = r_param: mathematically unused in forward */
  const float* wmem  = (const float*)d_in[13];

  const size_t QKV = (size_t)BATCH * NUM_HEADS * SEQ_T * HEAD_DIM;  // 2,097,152 floats
  float* Qw = (float*)d_ws;
  float* Kw = Qw + QKV;
  float* Vw = Kw + QKV;
  float* AO = Vw + QKV;  // pre-Wo attention output, [B,T,D]

  float* out   = (float*)d_out;
  float* probs = out + (size_t)BATCH * SEQ_T * D_MODEL;

  dim3 gblk(128);                          // 4 waves: 16 rows x 64 cols per block
  dim3 ggrd(D_MODEL / 64, M_ROWS / 16);    // (16, 128)

  gemm_bias_kernel<true><<<ggrd, gblk, 0, stream>>>(query, Wq, bq, Qw);
  gemm_bias_kernel<true><<<ggrd, gblk, 0, stream>>>(key,   Wk, bk, Kw);
  gemm_bias_kernel<true><<<ggrd, gblk, 0, stream>>>(value, Wv, bv, Vw);

  attention_kernel<<<dim3(SEQ_T / 16, BATCH * NUM_HEADS), 256, 0, stream>>>(
      Qw, Kw, Vw, tau, wmem, probs, AO);

  gemm_bias_kernel<false><<<ggrd, gblk, 0, stream>>>(AO, Wo, bo, out);
}